// ChamferLoss_12584254177838
// MI455X (gfx1250) — compile-verified
//
#include <hip/hip_runtime.h>

typedef __attribute__((ext_vector_type(2))) float v2f;
typedef __attribute__((ext_vector_type(8))) float v8f;

#define BB 2
#define NN 8192
#define MM 8192
#define ROWS_PER_WAVE 128   // 8 row-tiles of 16
#define COLS_PER_WAVE 512   // 32 col-steps of 16

// ---------------------------------------------------------------------------
// Init workspace: colpack = (d2bits<<32 | rowidx) running min, rowminU = +inf
// ---------------------------------------------------------------------------
__global__ void init_ws_kernel(unsigned long long* __restrict__ colpack,
                               unsigned* __restrict__ rowminU) {
  int i = blockIdx.x * blockDim.x + threadIdx.x;
  if (i < BB * MM) colpack[i] = ~0ULL;
  if (i < BB * NN) rowminU[i] = 0x7F800000u;   // +inf bits
}

// ---------------------------------------------------------------------------
// Main kernel: one wave per block. d2 tiles via V_WMMA_F32_16X16X4_F32.
//   A row (pred n):  (-2px, -2py, -2pz, |p|^2)   K0,K1 lanes 0-15; K2,K3 lanes 16-31
//   B col (ref  m):  ( rx,   ry,   rz,  1   )    same K split
//   acc = |p|^2 - 2 p.r ;  d2 = max(acc + |r|^2, 0)
// C/D layout: lane L holds column (L&15); VGPR v holds row v (+8 for L>=16).
// ---------------------------------------------------------------------------
__global__ __launch_bounds__(32) void chamfer_main(
    const float* __restrict__ pred_pts, const float* __restrict__ ref_pts,
    unsigned long long* __restrict__ colpack, unsigned* __restrict__ rowminU)
{
  const int  lane = threadIdx.x;
  const int  r16  = lane & 15;
  const bool lo   = lane < 16;
  const int  b    = blockIdx.z;
  const int  rowBase = blockIdx.y * ROWS_PER_WAVE;
  const int  colBase = blockIdx.x * COLS_PER_WAVE;
  const int  rowOff  = lo ? 0 : 8;

  v2f   A[8];
  float rowAcc[8][8];
  #pragma unroll
  for (int t = 0; t < 8; ++t) {
    const int row = rowBase + t * 16 + r16;
    const float* p = pred_pts + ((size_t)b * NN + row) * 3;
    const float px = p[0], py = p[1], pz = p[2];
    const float pn = px * px + py * py + pz * pz;
    A[t].x = lo ? (-2.0f * px) : (-2.0f * pz);
    A[t].y = lo ? (-2.0f * py) : pn;
    #pragma unroll
    for (int v = 0; v < 8; ++v) rowAcc[t][v] = 3.402823466e+38f;
  }

  const float* refBase = ref_pts + (size_t)b * MM * 3;

  // Software-pipelined ref-point load (B tile for next step)
  float rx, ry, rz;
  {
    const float* r = refBase + (size_t)(colBase + r16) * 3;
    rx = r[0]; ry = r[1]; rz = r[2];
  }

  for (int step = 0; step < COLS_PER_WAVE / 16; ++step) {
    const int col = colBase + step * 16 + r16;
    const float crx = rx, cry = ry, crz = rz;
    if (step + 1 < COLS_PER_WAVE / 16) {           // prefetch next step's point
      const float* r = refBase + (size_t)(col + 16) * 3;
      rx = r[0]; ry = r[1]; rz = r[2];
    }
    const float rn = crx * crx + cry * cry + crz * crz;
    v2f Bv;
    Bv.x = lo ? crx : crz;
    Bv.y = lo ? cry : 1.0f;

    // 8 independent per-tile argmin chains (breaks the 64-deep serial chain)
    float tv[8];
    int   ti[8];
    #pragma unroll
    for (int t = 0; t < 8; ++t) {
      v8f c = {};
      c = __builtin_amdgcn_wmma_f32_16x16x4_f32(false, A[t], false, Bv,
                                                (short)0, c, false, false);
      float bv = 3.402823466e+38f;
      int   bi = 0;
      #pragma unroll
      for (int v = 0; v < 8; ++v) {
        float d = fmaxf(c[v] + rn, 0.0f);
        rowAcc[t][v] = fminf(rowAcc[t][v], d);
        if (d < bv) { bv = d; bi = v; }            // v ascending: first-occurrence
      }
      tv[t] = bv;
      ti[t] = rowBase + t * 16 + rowOff + bi;
    }
    // Tree merge; left operand always has strictly smaller row indices, so
    // strict '<' keeps first occurrence on ties.
    #pragma unroll
    for (int s = 1; s < 8; s <<= 1) {
      #pragma unroll
      for (int t = 0; t < 8; t += 2 * s) {
        if (tv[t + s] < tv[t]) { tv[t] = tv[t + s]; ti[t] = ti[t + s]; }
      }
    }
    float cval = tv[0];
    int   cidx = ti[0];

    // merge lane pair (L, L^16): rows of the two halves interleave, so tie
    // needs the explicit smaller-index rule here.
    const float oval = __shfl_xor(cval, 16, 32);
    const int   oidx = __shfl_xor(cidx, 16, 32);
    if (oval < cval || (oval == cval && oidx < cidx)) { cval = oval; cidx = oidx; }

    const unsigned long long pk =
        ((unsigned long long)__float_as_uint(cval) << 32) |
        (unsigned long long)(unsigned)cidx;
    if (lo) atomicMin(&colpack[(size_t)b * MM + col], pk);
  }

  // Reduce row minima over the 16 column-lanes of each half, then publish.
  #pragma unroll
  for (int t = 0; t < 8; ++t) {
    #pragma unroll
    for (int v = 0; v < 8; ++v) {
      float x = rowAcc[t][v];
      x = fminf(x, __shfl_xor(x, 1, 32));
      x = fminf(x, __shfl_xor(x, 2, 32));
      x = fminf(x, __shfl_xor(x, 4, 32));
      x = fminf(x, __shfl_xor(x, 8, 32));
      rowAcc[t][v] = x;
    }
  }
  if (lane == 0 || lane == 16) {
    #pragma unroll
    for (int t = 0; t < 8; ++t)
      #pragma unroll
      for (int v = 0; v < 8; ++v)
        atomicMin(&rowminU[(size_t)b * NN + rowBase + t * 16 + v + rowOff],
                  __float_as_uint(rowAcc[t][v]));   // valid: d2 >= 0
  }
}

// ---------------------------------------------------------------------------
// Finalize: deterministic single-block reduction + gathers -> 3 outputs
// ---------------------------------------------------------------------------
__global__ __launch_bounds__(256) void finalize_kernel(
    const float* __restrict__ pred_sdfs, const float* __restrict__ pred_cols,
    const float* __restrict__ ref_sdfs,  const float* __restrict__ ref_cols,
    const unsigned long long* __restrict__ colpack,
    const unsigned* __restrict__ rowminU,
    float* __restrict__ out)
{
  __shared__ float s_sdf[256], s_col[256], s_cmin[256], s_rmin[256];
  const int tid = threadIdx.x;
  float a_sdf = 0.f, a_col = 0.f, a_cmin = 0.f, a_rmin = 0.f;

  for (int i = tid; i < BB * MM; i += 256) {
    const int b = i / MM, m = i % MM;
    const unsigned long long pk = colpack[i];
    const unsigned n = (unsigned)(pk & 0xFFFFFFFFu);
    a_cmin += __uint_as_float((unsigned)(pk >> 32));
    a_sdf  += fabsf(ref_sdfs[(size_t)b * MM + n] - pred_sdfs[(size_t)b * NN + m]);
    #pragma unroll
    for (int k = 0; k < 3; ++k)
      a_col += fabsf(ref_cols[((size_t)b * MM + n) * 3 + k] -
                     pred_cols[((size_t)b * NN + m) * 3 + k]);
  }
  for (int i = tid; i < BB * NN; i += 256)
    a_rmin += __uint_as_float(rowminU[i]);

  s_sdf[tid] = a_sdf; s_col[tid] = a_col; s_cmin[tid] = a_cmin; s_rmin[tid] = a_rmin;
  __syncthreads();
  for (int s = 128; s > 0; s >>= 1) {
    if (tid < s) {
      s_sdf[tid]  += s_sdf[tid + s];
      s_col[tid]  += s_col[tid + s];
      s_cmin[tid] += s_cmin[tid + s];
      s_rmin[tid] += s_rmin[tid + s];
    }
    __syncthreads();
  }
  if (tid == 0) {
    out[0] = s_sdf[0]  / (float)(BB * MM);
    out[1] = s_col[0]  / (float)(BB * MM * 3);
    out[2] = s_rmin[0] / (float)(BB * NN) + s_cmin[0] / (float)(BB * MM);
  }
}

// ---------------------------------------------------------------------------
extern "C" void kernel_launch(void* const* d_in, const int* in_sizes, int n_in,
                              void* d_out, int out_size, void* d_ws, size_t ws_size,
                              hipStream_t stream) {
  (void)in_sizes; (void)n_in; (void)out_size; (void)ws_size;
  const float* pred_pts  = (const float*)d_in[0];
  const float* pred_sdfs = (const float*)d_in[1];
  const float* pred_cols = (const float*)d_in[2];
  const float* ref_pts   = (const float*)d_in[3];
  const float* ref_sdfs  = (const float*)d_in[4];
  const float* ref_cols  = (const float*)d_in[5];

  unsigned long long* colpack = (unsigned long long*)d_ws;
  unsigned* rowminU =
      (unsigned*)((char*)d_ws + sizeof(unsigned long long) * (size_t)BB * MM);

  init_ws_kernel<<<(BB * MM + 255) / 256, 256, 0, stream>>>(colpack, rowminU);

  dim3 grid(MM / COLS_PER_WAVE, NN / ROWS_PER_WAVE, BB);
  chamfer_main<<<grid, 32, 0, stream>>>(pred_pts, ref_pts, colpack, rowminU);

  finalize_kernel<<<1, 256, 0, stream>>>(pred_sdfs, pred_cols, ref_sdfs, ref_cols,
                                         colpack, rowminU, (float*)d_out);
}